// MatchingPropagator_42004780155536
// MI455X (gfx1250) — compile-verified
//
#include <hip/hip_runtime.h>
#include <stdint.h>

// ---------------------------------------------------------------------------
// Problem constants (from reference: B=4, H=W=64)
// ---------------------------------------------------------------------------
#define Hdim   64
#define Wdim   64
#define Bdim   4
#define NPIX   4096      // pixels per image (H*W)
#define NTOT   16384     // B*H*W
#define PXPT   4         // pixels per thread
#define THREADS 1024     // 32 wave32 waves per workgroup

// CDNA5 async global->LDS path (gfx1250), guarded so host pass / older
// toolchains fall back to plain load+ds_store.
#if defined(__AMDGCN__) && __has_builtin(__builtin_amdgcn_global_load_async_to_lds_b32)
#define PM_HAVE_ASYNC 1
#define PM_AS1 __attribute__((address_space(1)))
#define PM_AS3 __attribute__((address_space(3)))
#else
#define PM_HAVE_ASYNC 0
#endif

// ---------------------------------------------------------------------------
// threefry2x32 (exact JAX PRNG core): 20 rounds, key injections every 4.
// ---------------------------------------------------------------------------
#define TF_ROUND(r) do { x0 += x1; x1 = (x1 << (r)) | (x1 >> (32 - (r))); x1 ^= x0; } while (0)

__host__ __device__ __forceinline__
void tf2x32(uint32_t k0, uint32_t k1, uint32_t c0, uint32_t c1,
            uint32_t& o0, uint32_t& o1) {
  uint32_t k2 = k0 ^ k1 ^ 0x1BD11BDAu;
  uint32_t x0 = c0 + k0;
  uint32_t x1 = c1 + k1;
  TF_ROUND(13); TF_ROUND(15); TF_ROUND(26); TF_ROUND(6);
  x0 += k1; x1 += k2 + 1u;
  TF_ROUND(17); TF_ROUND(29); TF_ROUND(16); TF_ROUND(24);
  x0 += k2; x1 += k0 + 2u;
  TF_ROUND(13); TF_ROUND(15); TF_ROUND(26); TF_ROUND(6);
  x0 += k0; x1 += k1 + 3u;
  TF_ROUND(17); TF_ROUND(29); TF_ROUND(16); TF_ROUND(24);
  x0 += k1; x1 += k2 + 4u;
  TF_ROUND(13); TF_ROUND(15); TF_ROUND(26); TF_ROUND(6);
  x0 += k2; x1 += k0 + 5u;
  o0 = x0; o1 = x1;
}

// Giles single-precision erfinv (what XLA lowers jnp erfinv to for f32).
__device__ __forceinline__ float pm_erfinv(float x) {
  float w = -log1pf(-x * x);
  float p;
  if (w < 5.0f) {
    w -= 2.5f;
    p = 2.81022636e-08f;
    p = fmaf(p, w, 3.43273939e-07f);
    p = fmaf(p, w, -3.5233877e-06f);
    p = fmaf(p, w, -4.39150654e-06f);
    p = fmaf(p, w, 0.00021858087f);
    p = fmaf(p, w, -0.00125372503f);
    p = fmaf(p, w, -0.00417768164f);
    p = fmaf(p, w, 0.246640727f);
    p = fmaf(p, w, 1.50140941f);
  } else {
    w = sqrtf(w) - 3.0f;
    p = -0.000200214257f;
    p = fmaf(p, w, 0.000100950558f);
    p = fmaf(p, w, 0.00134934322f);
    p = fmaf(p, w, -0.00367342844f);
    p = fmaf(p, w, 0.00573950773f);
    p = fmaf(p, w, -0.0076224613f);
    p = fmaf(p, w, 0.00943887047f);
    p = fmaf(p, w, 1.00167406f);
    p = fmaf(p, w, 2.83297682f);
  }
  return p * x;
}

// jax.random.normal element f of a (4,64,64,2) draw (32768 elems):
// counters split into halves [0..16383] | [16384..32767]; element f<16384 is
// x0-output of pair (f, f+16384), else x1-output of pair (f-16384, f).
__device__ __forceinline__ float pm_normal(uint32_t k0, uint32_t k1, uint32_t f) {
  uint32_t o0, o1, bits;
  if (f < 16384u) { tf2x32(k0, k1, f, f + 16384u, o0, o1); bits = o0; }
  else            { tf2x32(k0, k1, f - 16384u, f, o0, o1); bits = o1; }
  // uniform in [-1+2^-24, 1): bits -> [1,2) mantissa trick -> affine map
  float f01 = __uint_as_float((bits >> 9) | 0x3f800000u) - 1.0f;
  float u = fmaf(f01, 1.99999994f, -0.99999994f);
  u = fmaxf(u, -0.99999994f);
  return 1.41421356237f * pm_erfinv(u);   // sqrt(2)*erfinv(u)
}

__device__ __forceinline__ int imin(int a, int b) { return a < b ? a : b; }
__device__ __forceinline__ int imax(int a, int b) { return a > b ? a : b; }
__device__ __forceinline__ float clamp63(float v) { return fminf(fmaxf(v, 0.0f), 63.0f); }

// Bilinear score into this pixel's 64x64 correlation slice.
// fwd: base=p*4096, sy=64, sx=1 ; bwd: base=b*2^24+p_local, sy=262144, sx=4096
__device__ __forceinline__
float score_at(const float* __restrict__ corr, size_t base, int sy, int sx,
               float x, float y) {
  float x0f = floorf(x), y0f = floorf(y);
  float wx = x - x0f, wy = y - y0f;
  int x0 = imin(imax((int)x0f, 0), 63);
  int y0 = imin(imax((int)y0f, 0), 63);
  int x1 = imin(x0 + 1, 63);
  int y1 = imin(y0 + 1, 63);
  const float* pb = corr + base;
  float v00 = pb[y0 * sy + x0 * sx];
  float v01 = pb[y0 * sy + x1 * sx];
  float v10 = pb[y1 * sy + x0 * sx];
  float v11 = pb[y1 * sy + x1 * sx];
  float mx = 1.0f - wx, my = 1.0f - wy;
  return (v00 * mx + v01 * wx) * my + (v10 * mx + v11 * wx) * wy;
}

struct PMKeys { uint32_t k[12]; };  // [dir][step][hi/lo]

// ---------------------------------------------------------------------------
// Whole _handle() fused: one workgroup per (direction, batch image).
// Coordinate field lives in LDS (ping-pong), workgroup barriers between the
// 7 dependent steps. 8 blocks x 1024 threads, each thread owns 4 pixels.
// ---------------------------------------------------------------------------
__global__ __launch_bounds__(THREADS)
void pm_handle_kernel(const float* __restrict__ mf, const float* __restrict__ mb,
                      const float* __restrict__ corr,
                      float* __restrict__ resf, float* __restrict__ resb,
                      PMKeys keys) {
  __shared__ float sbx[2][NPIX];   // x-coords, ping-pong
  __shared__ float sby[2][NPIX];   // y-coords, ping-pong

  const int tid = threadIdx.x;
  const int dir = blockIdx.x >> 2;      // 0=fwd, 1=bwd
  const int b   = blockIdx.x & 3;
  const float* mt = dir ? mb : mf;      // matching input (B,2,H,W)
  float* ro = dir ? resb : resf;        // result, pixel-major (B*H*W, 2)
  const int mbase = b * 2 * NPIX;
  const int sy = dir ? (Wdim * NPIX) : Wdim;
  const int sx = dir ? NPIX : 1;

  float cx[PXPT], cy[PXPT];

  // ---- Stage initial coords into LDS buffer 0 --------------------------
#if PM_HAVE_ASYNC
  #pragma unroll
  for (int k = 0; k < PXPT; ++k) {
    const int pl = tid + (k << 10);
    __builtin_amdgcn_global_load_async_to_lds_b32(
        (PM_AS1 int*)(mt + mbase + pl),        (PM_AS3 int*)&sbx[0][pl], 0, 0);
    __builtin_amdgcn_global_load_async_to_lds_b32(
        (PM_AS1 int*)(mt + mbase + NPIX + pl), (PM_AS3 int*)&sby[0][pl], 0, 0);
  }
  #if __has_builtin(__builtin_amdgcn_s_wait_asynccnt)
  __builtin_amdgcn_s_wait_asynccnt(0);
  #else
  asm volatile("s_wait_asynccnt 0" ::: "memory");
  #endif
  __syncthreads();
  #pragma unroll
  for (int k = 0; k < PXPT; ++k) {
    const int pl = tid + (k << 10);
    cx[k] = sbx[0][pl];
    cy[k] = sby[0][pl];
  }
#else
  #pragma unroll
  for (int k = 0; k < PXPT; ++k) {
    const int pl = tid + (k << 10);
    cx[k] = mt[mbase + pl];
    cy[k] = mt[mbase + NPIX + pl];
    sbx[0][pl] = cx[k];
    sby[0][pl] = cy[k];
  }
  __syncthreads();
#endif

  // ---- 4 propagate steps, first 3 followed by fused random search ------
  const int DX[4] = { 1, -1, -1,  1 };
  const int DY[4] = { 1, -1,  1, -1 };
  int cur = 0;

  #pragma unroll
  for (int step = 0; step < 4; ++step) {
    const int dx = DX[step], dy = DY[step];
    float nx[PXPT], ny[PXPT];

    #pragma unroll
    for (int k = 0; k < PXPT; ++k) {
      const int pl = tid + (k << 10);
      const int h = pl >> 6, w = pl & 63;
      const size_t base = dir ? (((size_t)b << 24) + (size_t)pl)
                              : ((size_t)((b << 12) + pl) << 12);

      float bx = cx[k], by = cy[k];
      float bs = score_at(corr, base, sy, sx, bx, by);

      { // horizontal candidate: roll along W by dx, then +(dx,0), clamp
        const int wsrc = (w - dx) & 63;
        const float hx = clamp63(sbx[cur][(h << 6) + wsrc] + (float)dx);
        const float hy = clamp63(sby[cur][(h << 6) + wsrc]);
        const float hs = score_at(corr, base, sy, sx, hx, hy);
        if (hs > bs) { bx = hx; by = hy; bs = hs; }
      }
      { // vertical candidate: roll along H by dy, then +(0,dy), clamp
        const int hsrc = (h - dy) & 63;
        const float vx = clamp63(sbx[cur][(hsrc << 6) + w]);
        const float vy = clamp63(sby[cur][(hsrc << 6) + w] + (float)dy);
        const float vs = score_at(corr, base, sy, sx, vx, vy);
        if (vs > bs) { bx = vx; by = vy; bs = vs; }
      }

      if (step < 3) { // random search with key k{1,2,3}
        const uint32_t K0 = keys.k[dir * 6 + step * 2 + 0];
        const uint32_t K1 = keys.k[dir * 6 + step * 2 + 1];
        const uint32_t pg = (uint32_t)((b << 12) + pl);
        const float n0 = pm_normal(K0, K1, 2u * pg);
        const float n1 = pm_normal(K0, K1, 2u * pg + 1u);
        const float rx = clamp63(fmaf(3.0f, n0, bx));   // R = 3.0
        const float ry = clamp63(fmaf(3.0f, n1, by));
        const float rs = score_at(corr, base, sy, sx, rx, ry);
        if (rs - bs > 0.0f) { bx = rx; by = ry; }       // UPD_MULT = 1.0
      }
      nx[k] = bx; ny[k] = by;
    }

    const int nxt = cur ^ 1;
    #pragma unroll
    for (int k = 0; k < PXPT; ++k) {
      const int pl = tid + (k << 10);
      sbx[nxt][pl] = nx[k];
      sby[nxt][pl] = ny[k];
      cx[k] = nx[k];
      cy[k] = ny[k];
    }
    __syncthreads();
    cur = nxt;
  }

  // ---- Emit (B*H*W, 2) pixel-major result ------------------------------
  #pragma unroll
  for (int k = 0; k < PXPT; ++k) {
    const int pl = tid + (k << 10);
    const int pg = (b << 12) + pl;
    ro[pg * 2 + 0] = cx[k];
    ro[pg * 2 + 1] = cy[k];
  }
}

// ---------------------------------------------------------------------------
// Forward/backward consistency check + output transpose to (B,2,H,W).
// ---------------------------------------------------------------------------
__global__ __launch_bounds__(256)
void pm_final_kernel(const float* __restrict__ resf, const float* __restrict__ resb,
                     const float* __restrict__ mf, float* __restrict__ out) {
  const int p = blockIdx.x * blockDim.x + threadIdx.x;
  if (p >= NTOT) return;
  const int b = p >> 12, pl = p & 4095;

  const float fx = resf[p * 2 + 0];
  const float fy = resf[p * 2 + 1];

  const float x0f = floorf(fx), y0f = floorf(fy);
  const float wx = fx - x0f, wy = fy - y0f;
  const int x0 = imin(imax((int)x0f, 0), 63);
  const int y0 = imin(imax((int)y0f, 0), 63);
  const int x1 = imin(x0 + 1, 63);
  const int y1 = imin(y0 + 1, 63);

  const float* rb = resb + ((size_t)b << 12) * 2;
  const float mx = 1.0f - wx, my = 1.0f - wy;
  const float w00 = mx * my, w01 = wx * my, w10 = mx * wy, w11 = wx * wy;
  const int i00 = ((y0 << 6) + x0) * 2, i01 = ((y0 << 6) + x1) * 2;
  const int i10 = ((y1 << 6) + x0) * 2, i11 = ((y1 << 6) + x1) * 2;

  const float ctrx = rb[i00] * w00 + rb[i01] * w01 + rb[i10] * w10 + rb[i11] * w11;
  const float ctry = rb[i00 + 1] * w00 + rb[i01 + 1] * w01 + rb[i10 + 1] * w10 + rb[i11 + 1] * w11;

  const float diff = fmaxf(fabsf(fx - ctrx), fabsf(fy - ctry));
  float ox = fx, oy = fy;
  if (diff > 0.01f) {                       // EPS = 0.01 -> fall back to input
    ox = mf[b * 8192 + pl];
    oy = mf[b * 8192 + 4096 + pl];
  }
  out[b * 8192 + pl]        = ox;
  out[b * 8192 + 4096 + pl] = oy;
}

// ---------------------------------------------------------------------------
// Host entry: derive JAX keys (pure integer math, deterministic) and launch.
// key(42)=(0,42); split -> kf,kb; split(·,3) -> k1,k2,k3 per direction.
// ---------------------------------------------------------------------------
extern "C" void kernel_launch(void* const* d_in, const int* in_sizes, int n_in,
                              void* d_out, int out_size, void* d_ws, size_t ws_size,
                              hipStream_t stream) {
  (void)in_sizes; (void)n_in; (void)out_size; (void)ws_size;
  const float* mf   = (const float*)d_in[0];   // matching_f (B,2,H,W)
  const float* mb   = (const float*)d_in[1];   // matching_b (B,2,H,W)
  const float* corr = (const float*)d_in[2];   // corr_map   (B,H,W,H,W)

  float* resf = (float*)d_ws;                  // (B*H*W, 2)
  float* resb = resf + NTOT * 2;               // (B*H*W, 2)

  // split(key(42), 2): counts [0,1,2,3] -> pairs (0,2),(1,3)
  uint32_t a0, b0, a1, b1;
  tf2x32(0u, 42u, 0u, 2u, a0, b0);
  tf2x32(0u, 42u, 1u, 3u, a1, b1);
  const uint32_t kd[2][2] = { { a0, a1 },   // kf
                              { b0, b1 } }; // kb

  PMKeys keys;
  for (int d = 0; d < 2; ++d) {
    // split(k, 3): counts [0..5] -> pairs (0,3),(1,4),(2,5)
    uint32_t A, D, B2, E, C, F;
    tf2x32(kd[d][0], kd[d][1], 0u, 3u, A, D);
    tf2x32(kd[d][0], kd[d][1], 1u, 4u, B2, E);
    tf2x32(kd[d][0], kd[d][1], 2u, 5u, C, F);
    keys.k[d * 6 + 0] = A;  keys.k[d * 6 + 1] = B2;  // k1
    keys.k[d * 6 + 2] = C;  keys.k[d * 6 + 3] = D;   // k2
    keys.k[d * 6 + 4] = E;  keys.k[d * 6 + 5] = F;   // k3
  }

  pm_handle_kernel<<<dim3(8), dim3(THREADS), 0, stream>>>(mf, mb, corr, resf, resb, keys);
  pm_final_kernel<<<dim3(NTOT / 256), dim3(256), 0, stream>>>(resf, resb, mf, (float*)d_out);
}